// Predictor_15582141350539
// MI455X (gfx1250) — compile-verified
//
#include <hip/hip_runtime.h>

#define HEADS 4
#define CH    32
#define HC    128
#define SLOPE 0.2f

typedef __attribute__((ext_vector_type(2))) float v2f;
typedef __attribute__((ext_vector_type(8))) float v8f;

// Float atomic-max via monotone integer reinterpretation (works across signs
// when the location is initialized to -inf).
__device__ __forceinline__ void atomic_max_f32(float* addr, float v) {
    if (v >= 0.0f) atomicMax((int*)addr, __float_as_int(v));
    else           atomicMin((unsigned int*)addr, __float_as_uint(v));
}

// ---------------------------------------------------------------------------
// out[M,NOUT] = A[M,128] @ W[128,NOUT] + bias[NOUT]   (NOUT compile-time)
//
// One wave32 per 16x64 output strip: 4 independent 16x16 accumulators fed by
// one shared A fragment per K-step -> 4-way independent WMMA chains (hides the
// WMMA->WMMA RAW hazard) and all loads use immediate offsets (NOUT constexpr).
//
// V_WMMA_F32_16X16X4_F32 fragments (wave32):
//   A (16x4): lanes 0-15 rows M, {K,K+1} in the v2f; lanes 16-31 hold {K+2,K+3}
//   B (4x16): lane holds column n0+(lane&15), rows K+2*(lane>>4)+{0,1}
//   C/D (16x16): VGPR r, lane l -> row r+8*(l>>4), col (l&15)
// ---------------------------------------------------------------------------
template <int NOUT>
__global__ __launch_bounds__(256) void gemm128_wmma(
        const float* __restrict__ A,
        const float* __restrict__ W,
        const float* __restrict__ bias,
        float* __restrict__ out,
        int M) {
    constexpr int GROUPS = NOUT / 64;               // 64-column groups
    const int totalTiles = (M >> 4) * GROUPS;
    const int tid  = blockIdx.x * (blockDim.x >> 5) + (threadIdx.x >> 5);
    if (tid >= totalTiles) return;

    const int tm   = tid / GROUPS;                  // 16-row tile index
    const int tg   = tid % GROUPS;                  // 64-col group index
    const int lane = threadIdx.x & 31;
    const int half = lane >> 4;                     // 0: lanes 0-15, 1: 16-31
    const int l    = lane & 15;
    const int n0   = tg * 64 + l;                   // this lane's base column

    const float* Abase = A + (size_t)(tm * 16 + l) * 128 + 2 * half;
    const float* Wbase = W + (size_t)(2 * half) * NOUT + n0;

    v8f c[4];
#pragma unroll
    for (int t = 0; t < 4; ++t) {
        float bv = bias[n0 + t * 16];               // bias pre-loaded into C
#pragma unroll
        for (int r = 0; r < 8; ++r) c[t][r] = bv;
    }

#pragma unroll
    for (int k0 = 0; k0 < 128; k0 += 4) {
        v2f a = *(const v2f*)(Abase + k0);          // 8B aligned, imm offset
#pragma unroll
        for (int t = 0; t < 4; ++t) {
            v2f b;                                  // imm offsets (NOUT const)
            b.x = Wbase[k0 * NOUT + t * 16];
            b.y = Wbase[k0 * NOUT + NOUT + t * 16];
            c[t] = __builtin_amdgcn_wmma_f32_16x16x4_f32(
                       /*neg_a=*/false, a, /*neg_b=*/false, b,
                       /*c_mod=*/(short)0, c[t],
                       /*reuse_a=*/false, /*reuse_b=*/false);
        }
    }

    float* Obase = out + (size_t)(tm * 16) * NOUT + n0;
#pragma unroll
    for (int t = 0; t < 4; ++t)
#pragma unroll
        for (int r = 0; r < 8; ++r)
            Obase[(size_t)(r + 8 * half) * NOUT + t * 16] = c[t][r];
}

// ---------------------------------------------------------------------------
// Zero accumulators, -inf the per-(node,head) running max, zero denominators.
// ---------------------------------------------------------------------------
__global__ void k_init(float* __restrict__ mmax, float* __restrict__ denom,
                       float* __restrict__ acc, int N) {
    int i = blockIdx.x * blockDim.x + threadIdx.x;
    if (i < N * HC) acc[i] = 0.0f;
    if (i < N * HEADS) { mmax[i] = -__builtin_inff(); denom[i] = 0.0f; }
}

// ---------------------------------------------------------------------------
// Per edge (one wave32 each): logits[e][h] = sum_c att[h][c] *
//   leaky_relu(xl[src][h][c] + xr[dst][h][c]); atomic running max per (dst,h).
// Edges e >= E are the appended self loops (src = dst = e - E).
// ---------------------------------------------------------------------------
__global__ void edge_logits(const long long* __restrict__ edges,
                            const float* __restrict__ xl,
                            const float* __restrict__ xr,
                            const float* __restrict__ att,
                            float* __restrict__ logits,
                            float* __restrict__ mmax,
                            int E, int EP) {
    const int wid  = blockIdx.x * (blockDim.x >> 5) + (threadIdx.x >> 5);
    const int lane = threadIdx.x & 31;
    if (wid >= EP) return;

    int s, d;
    if (wid < E) { s = (int)edges[wid]; d = (int)edges[(size_t)E + wid]; }
    else         { s = d = wid - E; }

    const float* xls = xl + (size_t)s * HC;
    const float* xrd = xr + (size_t)d * HC;

    float p[HEADS];
#pragma unroll
    for (int h = 0; h < HEADS; ++h) {
        float v = xls[h * CH + lane] + xrd[h * CH + lane];
        v = (v > 0.0f) ? v : SLOPE * v;
        p[h] = v * att[h * CH + lane];
    }
    // Reduce all 4 head partials across the 32 lanes simultaneously.
#pragma unroll
    for (int off = 16; off > 0; off >>= 1) {
#pragma unroll
        for (int h = 0; h < HEADS; ++h) p[h] += __shfl_xor(p[h], off, 32);
    }
    if (lane == 0) {
#pragma unroll
        for (int h = 0; h < HEADS; ++h) {
            logits[(size_t)wid * HEADS + h] = p[h];
            atomic_max_f32(&mmax[(size_t)d * HEADS + h], p[h]);
        }
    }
}

// ---------------------------------------------------------------------------
// Per edge: w[h] = exp(logit - max); denom[dst][h] += w[h];
// acc[dst][h][c] += w[h] * xl[src][h][c]   (unnormalized weighted sum).
// ---------------------------------------------------------------------------
__global__ void edge_accum(const long long* __restrict__ edges,
                           const float* __restrict__ xl,
                           const float* __restrict__ logits,
                           const float* __restrict__ mmax,
                           float* __restrict__ denom,
                           float* __restrict__ acc,
                           int E, int EP) {
    const int wid  = blockIdx.x * (blockDim.x >> 5) + (threadIdx.x >> 5);
    const int lane = threadIdx.x & 31;
    if (wid >= EP) return;

    int s, d;
    if (wid < E) { s = (int)edges[wid]; d = (int)edges[(size_t)E + wid]; }
    else         { s = d = wid - E; }

    float w[HEADS];
#pragma unroll
    for (int h = 0; h < HEADS; ++h)
        w[h] = __expf(logits[(size_t)wid * HEADS + h] -
                      mmax[(size_t)d * HEADS + h]);

    if (lane == 0) {
#pragma unroll
        for (int h = 0; h < HEADS; ++h)
            atomicAdd(&denom[(size_t)d * HEADS + h], w[h]);
    }

    const float* xls  = xl  + (size_t)s * HC;
    float*       accd = acc + (size_t)d * HC;
#pragma unroll
    for (int h = 0; h < HEADS; ++h)
        atomicAdd(&accd[h * CH + lane], w[h] * xls[h * CH + lane]);
}

// ---------------------------------------------------------------------------
// f = relu(acc/denom + bias)            (residual == 0)
// f = f + 0.5 * relu(acc/denom + bias)  (residual == 1)
// ---------------------------------------------------------------------------
__global__ void k_finalize(const float* __restrict__ acc,
                           const float* __restrict__ denom,
                           const float* __restrict__ bias,
                           float* __restrict__ f,
                           int N, int residual) {
    int i = blockIdx.x * blockDim.x + threadIdx.x;
    if (i >= N * HC) return;
    int node = i >> 7;
    int col  = i & (HC - 1);
    int h    = col >> 5;
    float v = acc[i] / denom[node * HEADS + h] + bias[col];
    v = fmaxf(v, 0.0f);
    if (residual) f[i] += 0.5f * v;
    else          f[i] = v;
}

// ---------------------------------------------------------------------------
extern "C" void kernel_launch(void* const* d_in, const int* in_sizes, int n_in,
                              void* d_out, int out_size, void* d_ws, size_t ws_size,
                              hipStream_t stream) {
    const float*      data  = (const float*)d_in[0];
    const long long*  edges = (const long long*)d_in[1];
    const float* Wl1 = (const float*)d_in[2];
    const float* bl1 = (const float*)d_in[3];
    const float* Wr1 = (const float*)d_in[4];
    const float* br1 = (const float*)d_in[5];
    const float* at1 = (const float*)d_in[6];
    const float* b1  = (const float*)d_in[7];
    const float* Wl2 = (const float*)d_in[8];
    const float* bl2 = (const float*)d_in[9];
    const float* Wr2 = (const float*)d_in[10];
    const float* br2 = (const float*)d_in[11];
    const float* at2 = (const float*)d_in[12];
    const float* b2  = (const float*)d_in[13];
    const float* Wd  = (const float*)d_in[14];
    const float* bd  = (const float*)d_in[15];

    const int N  = in_sizes[0] / HC;   // 50000
    const int E  = in_sizes[1] / 2;    // 800000
    const int EP = E + N;              // edges + self loops

    // Workspace carve-out (256B aligned slabs).
    char* ws = (char*)d_ws;
    size_t off = 0;
    auto carve = [&](size_t bytes) {
        void* p = ws + off;
        off += (bytes + 255) & ~(size_t)255;
        return p;
    };
    float* f      = (float*)carve((size_t)N * HC * sizeof(float));
    float* xl     = (float*)carve((size_t)N * HC * sizeof(float));
    float* xr     = (float*)carve((size_t)N * HC * sizeof(float));
    float* acc    = (float*)carve((size_t)N * HC * sizeof(float));
    float* mmax   = (float*)carve((size_t)N * HEADS * sizeof(float));
    float* denom  = (float*)carve((size_t)N * HEADS * sizeof(float));
    float* logits = (float*)carve((size_t)EP * HEADS * sizeof(float));
    (void)ws_size; (void)n_in; (void)out_size;

    const int nodeBlocks = (N * HC + 255) / 256;
    const int edgeBlocks = (EP + 7) / 8;          // 8 waves per 256-thread block

    auto gat_layer = [&](const float* x, const float* Wl, const float* bl,
                         const float* Wr, const float* br, const float* att,
                         const float* bias, int residual) {
        const int tiles = (N >> 4) * (HC / 64);   // 16x64 strips
        gemm128_wmma<HC><<<(tiles + 7) / 8, 256, 0, stream>>>(x, Wl, bl, xl, N);
        gemm128_wmma<HC><<<(tiles + 7) / 8, 256, 0, stream>>>(x, Wr, br, xr, N);
        k_init<<<nodeBlocks, 256, 0, stream>>>(mmax, denom, acc, N);
        edge_logits<<<edgeBlocks, 256, 0, stream>>>(edges, xl, xr, att, logits, mmax, E, EP);
        edge_accum<<<edgeBlocks, 256, 0, stream>>>(edges, xl, logits, mmax, denom, acc, E, EP);
        k_finalize<<<nodeBlocks, 256, 0, stream>>>(acc, denom, bias, f, N, residual);
    };

    gat_layer(data, Wl1, bl1, Wr1, br1, at1, b1, /*residual=*/0);
    for (int blk = 0; blk < 3; ++blk)
        gat_layer(f, Wl2, bl2, Wr2, br2, at2, b2, /*residual=*/1);

    // Dense head: out[N,64] = f @ Wd + bd
    const int tilesD = (N >> 4) * (64 / 64);
    gemm128_wmma<64><<<(tilesD + 7) / 8, 256, 0, stream>>>(f, Wd, bd, (float*)d_out, N);
}